// UFNO_SMM_15298673508726
// MI455X (gfx1250) — compile-verified
//
#include <hip/hip_runtime.h>
#include <hip/hip_bf16.h>

// ---------------------------------------------------------------------------
// UFNO on gfx1250. All GEMM-shaped contractions run on V_WMMA_F32_16X16X4_F32
// (fp32 matrix pipe -> complex64 accuracy). All operands are padded to
// multiples of 16 (M,N) / 4 (K) in workspace with zero fill, so the WMMA
// inner loop has ZERO bounds checks and EXEC stays all-ones trivially.
// Working set (~122 MB scratch, 37 MB live state) is L2-resident (192 MB).
// ---------------------------------------------------------------------------

constexpr int kSY   = 512;
constexpr int kSX   = 282;
constexpr int kSXp  = 288;          // padded x (mult of 16)
constexpr int kW    = 32;           // channels
constexpr int kM1   = 20;
constexpr int kM2   = 20;
constexpr int kM2p  = 32;           // padded x-modes
constexpr int kKY   = 40;           // y-modes (pos+neg)
constexpr int kKYp  = 48;           // padded y-modes
constexpr int kPix  = kSY * kSX;    // 144384 (compact, for I/O)
constexpr int kPixP = kSY * kSXp;   // 147456 (padded, internal)
constexpr int kModesP = kKYp * kM2p; // 1536 padded modes per channel

typedef float v2f __attribute__((ext_vector_type(2)));
typedef float v8f __attribute__((ext_vector_type(8)));

__device__ __forceinline__ v8f wmma_f32(v2f a, v2f b, v8f c) {
  // (neg_a, A, neg_b, B, c_mod, C, reuse_a, reuse_b)
  return __builtin_amdgcn_wmma_f32_16x16x4_f32(false, a, false, b, (short)0, c,
                                               false, false);
}

__device__ __forceinline__ float gelu_exact(float x) {
  return 0.5f * x * (1.0f + erff(x * 0.70710678118654752f));
}

// ---------------------------------------------------------------------------
// Vandermonde build, zero-padded: Vy [48][512] (rows 40..47 = 0),
// Vx [32][288] (rows 20..31 = 0, cols 282..287 = 0).
// ---------------------------------------------------------------------------
__global__ void k_vand(const float* __restrict__ xp, const float* __restrict__ yp,
                       float* __restrict__ Vyr, float* __restrict__ Vyi,
                       float* __restrict__ Vxr, float* __restrict__ Vxi) {
  int idx = blockIdx.x * blockDim.x + threadIdx.x;
  const int ny = kKYp * kSY;
  if (idx < ny) {
    int k = idx / kSY, y = idx % kSY;
    float re = 0.0f, im = 0.0f;
    if (k < kKY) {
      float ky = (k < kM1) ? (float)k : (float)(k - kKY);
      float a = ky * yp[y];
      const float invs = 1.0f / sqrtf((float)kSY);
      re = cosf(a) * invs;
      im = -sinf(a) * invs;
    }
    Vyr[idx] = re;
    Vyi[idx] = im;
  } else {
    int j = idx - ny;
    if (j < kM2p * kSXp) {
      int l = j / kSXp, x = j % kSXp;
      float re = 0.0f, im = 0.0f;
      if (l < kM2 && x < kSX) {
        float a = (float)l * xp[x];
        const float invs = 1.0f / sqrtf((float)kSX);
        re = cosf(a) * invs;
        im = -sinf(a) * invs;
      }
      Vxr[j] = re;
      Vxi[j] = im;
    }
  }
}

// ---------------------------------------------------------------------------
// Lift into padded h [32][512][288]; pad columns written as zero.
// ---------------------------------------------------------------------------
__global__ void k_lift(const float* __restrict__ xin,
                       const float2* __restrict__ fc0_w,
                       const float2* __restrict__ fc0_b,
                       float* __restrict__ hr, float* __restrict__ hi) {
  int idx = blockIdx.x * blockDim.x + threadIdx.x;
  if (idx >= kPixP) return;
  int y = idx / kSXp, x = idx % kSXp;
  if (x >= kSX) {
#pragma unroll 4
    for (int c = 0; c < kW; ++c) {
      hr[c * kPixP + idx] = 0.0f;
      hi[c * kPixP + idx] = 0.0f;
    }
    return;
  }
  float xv = xin[y * kSX + x];
  float gx = (float)x * (1.0f / (float)(kSX - 1));
  float gy = (float)y * (1.0f / (float)(kSY - 1));
#pragma unroll 4
  for (int c = 0; c < kW; ++c) {
    float2 w0 = fc0_w[c], w1 = fc0_w[kW + c], w2 = fc0_w[2 * kW + c];
    float2 b = fc0_b[c];
    hr[c * kPixP + idx] = fmaf(xv, w0.x, fmaf(gx, w1.x, fmaf(gy, w2.x, b.x)));
    hi[c * kPixP + idx] = fmaf(xv, w0.y, fmaf(gx, w1.y, fmaf(gy, w2.y, b.y)));
  }
}

// ---------------------------------------------------------------------------
// Guard-free batched strided GEMM on fp32 WMMA.
//   CPLX=true : C = A*B complex (4 wmma/k-step), conj via sign at epilogue.
//   CPLX=false: Cr = Ar*Br, Ci = Ai*Bi (pointwise layer, 2 wmma/k-step).
//   AUNIT=true: A k-stride == 1 -> 64-bit A fragment loads.
// Requires: M,N multiples of 16; K multiple of 4; operands zero-padded.
// ---------------------------------------------------------------------------
template <bool AUNIT, bool CPLX>
__device__ __forceinline__ void gemm_body(
    const float* __restrict__ apr, const float* __restrict__ api,
    const float* __restrict__ bpr, const float* __restrict__ bpi,
    int K, int sAk, int sBk, int half,
    v8f& rr, v8f& ii, v8f& ri, v8f& ir) {
  for (int kb = 0; kb < K; kb += 4) {
    int k0 = kb + 2 * half;
    int k1 = k0 + 1;
    v2f fAr, fAi, fBr, fBi;
    if (AUNIT) {
      fAr = *(const v2f*)(apr + k0);
      fAi = *(const v2f*)(api + k0);
    } else {
      fAr.x = apr[k0 * sAk];
      fAr.y = apr[k1 * sAk];
      fAi.x = api[k0 * sAk];
      fAi.y = api[k1 * sAk];
    }
    fBr.x = bpr[k0 * sBk];
    fBr.y = bpr[k1 * sBk];
    fBi.x = bpi[k0 * sBk];
    fBi.y = bpi[k1 * sBk];
    rr = wmma_f32(fAr, fBr, rr);
    ii = wmma_f32(fAi, fBi, ii);
    if (CPLX) {
      ri = wmma_f32(fAr, fBi, ri);
      ir = wmma_f32(fAi, fBr, ir);
    }
  }
}

__global__ __launch_bounds__(256) void k_cgemm(
    const float* __restrict__ Ar, const float* __restrict__ Ai,
    const float* __restrict__ Br, const float* __restrict__ Bi,
    float* __restrict__ Cr, float* __restrict__ Ci,
    int M, int N, int K,
    int sAm, int sAk, int bA,
    int sBk, int sBn, int bB,
    int sCm, int sCn, int bC,
    int nbatch, int conjA, int conjB, int mode) {
  int tilesM = M >> 4, tilesN = N >> 4;
  int tilesPer = tilesM * tilesN;
  int wave = blockIdx.x * (blockDim.x >> 5) + (threadIdx.x >> 5);
  if (wave >= tilesPer * nbatch) return;  // wave-uniform exit
  int lane = threadIdx.x & 31;
  int bat = wave / tilesPer;
  int t = wave - bat * tilesPer;
  int tm = (t / tilesN) << 4;
  int tn = (t - (t / tilesN) * tilesN) << 4;

  int half = lane >> 4;  // 0/1
  int lm = lane & 15;
  int m = tm + lm;
  int n = tn + lm;

  const float* apr = Ar + (size_t)bat * bA + (size_t)m * sAm;
  const float* api = Ai + (size_t)bat * bA + (size_t)m * sAm;
  const float* bpr = Br + (size_t)bat * bB + (size_t)n * sBn;
  const float* bpi = Bi + (size_t)bat * bB + (size_t)n * sBn;
  float* cr = Cr + (size_t)bat * bC;
  float* ci = Ci + (size_t)bat * bC;

  v8f rr = {}, ii = {}, ri = {}, ir = {};

  if (mode == 0) {
    if (sAk == 1) gemm_body<true, true>(apr, api, bpr, bpi, K, sAk, sBk, half, rr, ii, ri, ir);
    else          gemm_body<false, true>(apr, api, bpr, bpi, K, sAk, sBk, half, rr, ii, ri, ir);
  } else {
    if (sAk == 1) gemm_body<true, false>(apr, api, bpr, bpi, K, sAk, sBk, half, rr, ii, ri, ir);
    else          gemm_body<false, false>(apr, api, bpr, bpi, K, sAk, sBk, half, rr, ii, ri, ir);
  }

  // conj folded into epilogue signs: Cr = rr - sA*sB*ii ; Ci = sB*ri + sA*ir
  float sA = conjA ? -1.0f : 1.0f;
  float sB = conjB ? -1.0f : 1.0f;
  float sAB = sA * sB;
#pragma unroll
  for (int v = 0; v < 8; ++v) {
    int mm = tm + v + 8 * half;
    int nn = tn + lm;
    size_t co = (size_t)mm * sCm + (size_t)nn * sCn;
    float pr, pi;
    if (mode == 0) {
      pr = fmaf(-sAB, ii[v], rr[v]);
      pi = fmaf(sB, ri[v], sA * ir[v]);
    } else {
      pr = rr[v];
      pi = ii[v];
    }
    cr[co] = pr;
    ci[co] = pi;
  }
}

// ---------------------------------------------------------------------------
// Per-mode channel mix on padded mode grid [32][48][32]; pad modes -> 0.
// o[o][k][l] = sum_i uf[i][k][l] * w[i][o][(k|k-M1)][l]
// ---------------------------------------------------------------------------
__global__ void k_modemix(const float* __restrict__ ufr, const float* __restrict__ ufi,
                          const float2* __restrict__ w1, const float2* __restrict__ w2,
                          float* __restrict__ outr, float* __restrict__ outi) {
  int idx = blockIdx.x * blockDim.x + threadIdx.x;
  if (idx >= kW * kModesP) return;
  int o = idx / kModesP;
  int rem = idx - o * kModesP;
  int k = rem / kM2p, l = rem - k * kM2p;
  if (k >= kKY || l >= kM2) {
    outr[idx] = 0.0f;
    outi[idx] = 0.0f;
    return;
  }
  const float2* wp;
  int kk;
  if (k < kM1) { wp = w1; kk = k; } else { wp = w2; kk = k - kM1; }
  float sr = 0.0f, si = 0.0f;
#pragma unroll 4
  for (int i = 0; i < kW; ++i) {
    float ur = ufr[i * kModesP + rem];
    float ui = ufi[i * kModesP + rem];
    float2 w = wp[((i * kW + o) * kM1 + kk) * kM2 + l];
    sr = fmaf(ur, w.x, fmaf(-ui, w.y, sr));
    si = fmaf(ur, w.y, fmaf(ui, w.x, si));
  }
  outr[idx] = sr;
  outi[idx] = si;
}

// ---------------------------------------------------------------------------
// Combine: h = h1 + h2 + bias(c), optional exact cgelu; pad columns -> 0.
// ---------------------------------------------------------------------------
__global__ void k_combine(const float* __restrict__ h1r, const float* __restrict__ h1i,
                          const float* __restrict__ h2r, const float* __restrict__ h2i,
                          const float* __restrict__ br, const float* __restrict__ bi,
                          float* __restrict__ hr, float* __restrict__ hi, int dogelu) {
  int idx = blockIdx.x * blockDim.x + threadIdx.x;
  if (idx >= kW * kPixP) return;
  int x = idx % kSXp;  // valid: kPixP and c*kPixP are multiples of kSXp
  if (x >= kSX) {
    hr[idx] = 0.0f;
    hi[idx] = 0.0f;
    return;
  }
  int c = idx / kPixP;
  float vr = h1r[idx] + h2r[idx] + br[c];
  float vi = h1i[idx] + h2i[idx] + bi[c];
  if (dogelu) { vr = gelu_exact(vr); vi = gelu_exact(vi); }
  hr[idx] = vr;
  hi[idx] = vi;
}

// ---------------------------------------------------------------------------
// Head: z = cgelu(h @ fc1_w + fc1_b); out = z @ fc2_w + fc2_b.
// fc1 weights staged in LDS (33 KB of the 320 KB WGP pool).
// ---------------------------------------------------------------------------
__global__ __launch_bounds__(256) void k_head(
    const float* __restrict__ hr, const float* __restrict__ hi,
    const float2* __restrict__ fc1_w, const float2* __restrict__ fc1_b,
    const float2* __restrict__ fc2_w, const float2* __restrict__ fc2_b,
    float2* __restrict__ out) {
  __shared__ float2 w1s[kW * 128];
  __shared__ float2 b1s[128];
  int tid = threadIdx.x;
  for (int i = tid; i < kW * 128; i += 256) w1s[i] = fc1_w[i];
  if (tid < 128) b1s[tid] = fc1_b[tid];
  __syncthreads();

  int idx = blockIdx.x * 256 + tid;
  if (idx >= kPix) return;
  int y = idx / kSX, x = idx - y * kSX;
  int src = y * kSXp + x;
  float xr[kW], xi[kW];
#pragma unroll
  for (int c = 0; c < kW; ++c) {
    xr[c] = hr[c * kPixP + src];
    xi[c] = hi[c * kPixP + src];
  }
  float accr = 0.0f, acci = 0.0f;
#pragma unroll 2
  for (int j = 0; j < 128; ++j) {
    float zr = b1s[j].x, zi = b1s[j].y;
#pragma unroll
    for (int c = 0; c < kW; ++c) {
      float2 w = w1s[c * 128 + j];
      zr = fmaf(xr[c], w.x, fmaf(-xi[c], w.y, zr));
      zi = fmaf(xr[c], w.y, fmaf(xi[c], w.x, zi));
    }
    zr = gelu_exact(zr);
    zi = gelu_exact(zi);
    float2 w2 = fc2_w[j];
    accr = fmaf(zr, w2.x, fmaf(-zi, w2.y, accr));
    acci = fmaf(zr, w2.y, fmaf(zi, w2.x, acci));
  }
  float2 b2 = fc2_b[0];
  out[idx] = make_float2(accr + b2.x, acci + b2.y);
}

// ---------------------------------------------------------------------------
// Host side
// ---------------------------------------------------------------------------
static void cgemm(hipStream_t s,
                  const float* Ar, const float* Ai, const float* Br, const float* Bi,
                  float* Cr, float* Ci, int M, int N, int K,
                  int sAm, int sAk, int bA, int sBk, int sBn, int bB,
                  int sCm, int sCn, int bC, int nb, int cA, int cB, int mode) {
  int tM = M / 16, tN = N / 16;
  long total = (long)tM * tN * nb;
  const int wpb = 8;  // 8 waves (256 threads) per block
  int blocks = (int)((total + wpb - 1) / wpb);
  k_cgemm<<<blocks, 32 * wpb, 0, s>>>(Ar, Ai, Br, Bi, Cr, Ci, M, N, K,
                                      sAm, sAk, bA, sBk, sBn, bB,
                                      sCm, sCn, bC, nb, cA, cB, mode);
}

extern "C" void kernel_launch(void* const* d_in, const int* in_sizes, int n_in,
                              void* d_out, int out_size, void* d_ws, size_t ws_size,
                              hipStream_t stream) {
  (void)in_sizes; (void)n_in; (void)out_size; (void)ws_size;
  const float*  x     = (const float*)d_in[0];
  const float*  xp    = (const float*)d_in[1];
  const float*  yp    = (const float*)d_in[2];
  const float2* fc0_w = (const float2*)d_in[3];
  const float2* fc0_b = (const float2*)d_in[4];
  const float2* sc_w1 = (const float2*)d_in[5];
  const float2* sc_w2 = (const float2*)d_in[6];
  const float*  pw_wr = (const float*)d_in[7];
  const float*  pw_wi = (const float*)d_in[8];
  const float*  pw_br = (const float*)d_in[9];
  const float*  pw_bi = (const float*)d_in[10];
  const float2* fc1_w = (const float2*)d_in[11];
  const float2* fc1_b = (const float2*)d_in[12];
  const float2* fc2_w = (const float2*)d_in[13];
  const float2* fc2_b = (const float2*)d_in[14];

  float* ws = (float*)d_ws;
  size_t off = 0;
  auto take = [&](size_t nflt) {
    float* p = ws + off;
    off += (nflt + 63) & ~(size_t)63;  // 256B align
    return p;
  };
  float* Vyr = take((size_t)kKYp * kSY);
  float* Vyi = take((size_t)kKYp * kSY);
  float* Vxr = take((size_t)kM2p * kSXp);
  float* Vxi = take((size_t)kM2p * kSXp);
  float* hr  = take((size_t)kW * kPixP);
  float* hi  = take((size_t)kW * kPixP);
  float* h1r = take((size_t)kW * kPixP);
  float* h1i = take((size_t)kW * kPixP);
  float* h2r = take((size_t)kW * kPixP);
  float* h2i = take((size_t)kW * kPixP);
  float* tr  = take((size_t)kW * kKYp * kSXp);
  float* ti  = take((size_t)kW * kKYp * kSXp);
  float* ufr = take((size_t)kW * kModesP);
  float* ufi = take((size_t)kW * kModesP);
  float* omr = take((size_t)kW * kModesP);
  float* omi = take((size_t)kW * kModesP);
  float* t2r = take((size_t)kW * kSY * kM2p);
  float* t2i = take((size_t)kW * kSY * kM2p);

  k_vand<<<(kKYp * kSY + kM2p * kSXp + 255) / 256, 256, 0, stream>>>(
      xp, yp, Vyr, Vyi, Vxr, Vxi);
  k_lift<<<(kPixP + 255) / 256, 256, 0, stream>>>(x, fc0_w, fc0_b, hr, hi);

  for (int L = 0; L < 6; ++L) {
    // (A) t[c,k,x] = sum_y Vy[k,y]*h[c,y,x]        M=48 N=288 K=512, batch 32
    cgemm(stream, Vyr, Vyi, hr, hi, tr, ti, kKYp, kSXp, kSY,
          kSY, 1, 0, kSXp, 1, kPixP, kSXp, 1, kKYp * kSXp, kW, 0, 0, 0);
    // (B) uf[c,k,l] = sum_x t[c,k,x]*Vx[l,x]       M=48 N=32 K=288
    cgemm(stream, tr, ti, Vxr, Vxi, ufr, ufi, kKYp, kM2p, kSXp,
          kSXp, 1, kKYp * kSXp, 1, kSXp, 0, kM2p, 1, kModesP, kW, 0, 0, 0);
    // (C) per-mode 32x32 channel mix (pad modes -> 0)
    k_modemix<<<(kW * kModesP + 255) / 256, 256, 0, stream>>>(
        ufr, ufi, sc_w1 + (size_t)L * kW * kW * kM1 * kM2,
        sc_w2 + (size_t)L * kW * kW * kM1 * kM2, omr, omi);
    // (D) t2[c,y,l] = sum_k conj(Vy[k,y])*o[c,k,l]  M=512 N=32 K=48 (A^T view)
    cgemm(stream, Vyr, Vyi, omr, omi, t2r, t2i, kSY, kM2p, kKYp,
          1, kSY, 0, kM2p, 1, kModesP, kM2p, 1, kSY * kM2p, kW, 1, 0, 0);
    // (E) h1[c,y,x] = sum_l t2[c,y,l]*conj(Vx[l,x]) M=512 N=288 K=32
    cgemm(stream, t2r, t2i, Vxr, Vxi, h1r, h1i, kSY, kSXp, kM2p,
          kM2p, 1, kSY * kM2p, kSXp, 1, 0, kSXp, 1, kPixP, kW, 0, 1, 0);
    // (F) pointwise split-real: h2.r = pw_wr@h.r ; h2.i = pw_wi@h.i
    cgemm(stream, pw_wr + (size_t)L * kW * kW, pw_wi + (size_t)L * kW * kW,
          hr, hi, h2r, h2i, kW, kPixP, kW,
          kW, 1, 0, kPixP, 1, 0, kPixP, 1, 0, 1, 0, 0, 1);
    // (G) h = h1 + h2 + bias, cgelu for layers 0..4; re-zero pad columns
    k_combine<<<(kW * kPixP + 255) / 256, 256, 0, stream>>>(
        h1r, h1i, h2r, h2i, pw_br + (size_t)L * kW, pw_bi + (size_t)L * kW,
        hr, hi, (L < 5) ? 1 : 0);
  }

  k_head<<<(kPix + 255) / 256, 256, 0, stream>>>(hr, hi, fc1_w, fc1_b, fc2_w,
                                                 fc2_b, (float2*)d_out);
}